// NeuroFusionGNN_53626961658419
// MI455X (gfx1250) — compile-verified
//
#include <hip/hip_runtime.h>
#include <math.h>

#define NN 20000
#define DD 256
#define NE 320000

typedef __attribute__((ext_vector_type(2))) float v2f;
typedef __attribute__((ext_vector_type(8))) float v8f;

__global__ void zero_f32(float* __restrict__ p, int n) {
  int i = blockIdx.x * blockDim.x + threadIdx.x;
  if (i < n) p[i] = 0.0f;
}

__global__ void degree_count(const int* __restrict__ dst, float* __restrict__ cnt) {
  int e = blockIdx.x * blockDim.x + threadIdx.x;
  if (e < NE) unsafeAtomicAdd(&cnt[dst[e]], 1.0f);
}

__global__ void inv_count(float* __restrict__ cnt) {
  int i = blockIdx.x * blockDim.x + threadIdx.x;
  if (i < NN) cnt[i] = 1.0f / fmaxf(cnt[i], 1.0f);
}

// one thread handles one edge x 8-float chunk: float4x2 gather + 8 f32 atomics
__global__ void scatter_add(const float* __restrict__ h, const int* __restrict__ src,
                            const int* __restrict__ dst, float* __restrict__ agg) {
  unsigned tid = blockIdx.x * blockDim.x + threadIdx.x;
  unsigned e = tid >> 5;
  if (e >= NE) return;
  unsigned c = (tid & 31u) << 3;
  int s = src[e], d = dst[e];
  const float4* p = (const float4*)(h + (size_t)s * DD + c);
  float4 v0 = p[0], v1 = p[1];
  float* q = agg + (size_t)d * DD + c;
  unsafeAtomicAdd(q + 0, v0.x); unsafeAtomicAdd(q + 1, v0.y);
  unsafeAtomicAdd(q + 2, v0.z); unsafeAtomicAdd(q + 3, v0.w);
  unsafeAtomicAdd(q + 4, v1.x); unsafeAtomicAdd(q + 5, v1.y);
  unsafeAtomicAdd(q + 6, v1.z); unsafeAtomicAdd(q + 7, v1.w);
}

// out = h + gelu(LN(mean @ Wl^T + bl + h @ Wr^T) * g + be)
// block = 16 node rows, 512 threads = 16 waves; wave w owns output cols [16w,16w+16)
__launch_bounds__(512, 2)
__global__ void sage_fused(const float* __restrict__ h,
                           const float* __restrict__ agg,
                           const float* __restrict__ invc,
                           const float* __restrict__ Wl,
                           const float* __restrict__ bl,
                           const float* __restrict__ Wr,
                           const float* __restrict__ g,
                           const float* __restrict__ be,
                           float* __restrict__ out) {
  __shared__ float lA[16][260];   // mean tile, padded stride: 260%64 = 4-bank row skew
  __shared__ float lX[16][260];   // raw input tile (GEMM A operand 2 + residual)
  __shared__ float rowsum[16];
  __shared__ float rowsq[16];

  const int tid = threadIdx.x;
  const int row0 = blockIdx.x * 16;

  if (tid < 16) { rowsum[tid] = 0.0f; rowsq[tid] = 0.0f; }

  { // stage tiles: 512 threads x 8 floats = 16x256 per tile
    int r = tid >> 5;
    int c0 = (tid & 31) << 3;
    int row = row0 + r;
    float ic = invc[row];
    const float4* ap = (const float4*)(agg + (size_t)row * DD + c0);
    const float4* xp = (const float4*)(h + (size_t)row * DD + c0);
    float4 a0 = ap[0], a1 = ap[1];
    float4 x0 = xp[0], x1 = xp[1];
    a0.x *= ic; a0.y *= ic; a0.z *= ic; a0.w *= ic;
    a1.x *= ic; a1.y *= ic; a1.z *= ic; a1.w *= ic;
    *(float4*)&lA[r][c0]     = a0;
    *(float4*)&lA[r][c0 + 4] = a1;
    *(float4*)&lX[r][c0]     = x0;
    *(float4*)&lX[r][c0 + 4] = x1;
  }
  __syncthreads();

  const int lane = tid & 31;
  const int wave = tid >> 5;
  const int col_base = wave * 16;
  const int mrow = lane & 15;
  const int ksel = (lane >> 4) << 1;     // f32 A/B tile: lanes 0-15 hold K,K+1; 16-31 hold K+2,K+3
  const int col = col_base + mrow;

  v8f acc = {};
  const float* la_p = &lA[mrow][ksel];
  const float* lx_p = &lX[mrow][ksel];
  const float* wl_p = Wl + (size_t)col * DD + ksel;  // B = W^T -> row-major W rows
  const float* wr_p = Wr + (size_t)col * DD + ksel;

#pragma unroll 8
  for (int k = 0; k < DD; k += 4) {
    v2f a = *(const v2f*)(la_p + k);
    v2f b = *(const v2f*)(wl_p + k);
    acc = __builtin_amdgcn_wmma_f32_16x16x4_f32(false, a, false, b, (short)0, acc,
                                                false, false);
  }
#pragma unroll 8
  for (int k = 0; k < DD; k += 4) {
    v2f a = *(const v2f*)(lx_p + k);
    v2f b = *(const v2f*)(wr_p + k);
    acc = __builtin_amdgcn_wmma_f32_16x16x4_f32(false, a, false, b, (short)0, acc,
                                                false, false);
  }

  float bb = bl[col];
#pragma unroll
  for (int r = 0; r < 8; ++r) acc[r] += bb;

  // per-row partial sums: acc[r] lane<16 -> row r, lane>=16 -> row r+8
#pragma unroll
  for (int r = 0; r < 8; ++r) {
    float s = acc[r];
    float q = s * s;
#pragma unroll
    for (int off = 1; off < 16; off <<= 1) {
      s += __shfl_xor(s, off, 32);
      q += __shfl_xor(q, off, 32);
    }
    if ((lane & 15) == 0) {
      int rl = r + ((lane >> 4) << 3);
      atomicAdd(&rowsum[rl], s);
      atomicAdd(&rowsq[rl], q);
    }
  }
  __syncthreads();

  float gg = g[col];
  float bbeta = be[col];
#pragma unroll
  for (int r = 0; r < 8; ++r) {
    int rl = r + ((lane >> 4) << 3);
    float mu = rowsum[rl] * (1.0f / 256.0f);
    float var = rowsq[rl] * (1.0f / 256.0f) - mu * mu;
    float rs = rsqrtf(var + 1e-5f);
    float hn = (acc[r] - mu) * rs * gg + bbeta;
    float ge = 0.5f * hn * (1.0f + erff(hn * 0.70710678118654752f));  // exact gelu
    out[(size_t)(row0 + rl) * DD + col] = lX[rl][col] + ge;
  }
}

extern "C" void kernel_launch(void* const* d_in, const int* in_sizes, int n_in,
                              void* d_out, int out_size, void* d_ws, size_t ws_size,
                              hipStream_t stream) {
  const float* x   = (const float*)d_in[0];
  const int*   ei  = (const int*)d_in[1];
  const float* W1l = (const float*)d_in[2];
  const float* b1l = (const float*)d_in[3];
  const float* W1r = (const float*)d_in[4];
  const float* g1  = (const float*)d_in[5];
  const float* be1 = (const float*)d_in[6];
  const float* W2l = (const float*)d_in[7];
  const float* b2l = (const float*)d_in[8];
  const float* W2r = (const float*)d_in[9];
  const float* g2  = (const float*)d_in[10];
  const float* be2 = (const float*)d_in[11];
  float* out = (float*)d_out;

  const int* src = ei;
  const int* dst = ei + NE;

  float* agg = (float*)d_ws;                       // N*D
  float* y1  = agg + (size_t)NN * DD;              // N*D
  float* cnt = y1 + (size_t)NN * DD;               // N

  const int ND = NN * DD;                          // 5,120,000

  zero_f32<<<(ND + 255) / 256, 256, 0, stream>>>(agg, ND);
  zero_f32<<<(NN + 255) / 256, 256, 0, stream>>>(cnt, NN);
  degree_count<<<(NE + 255) / 256, 256, 0, stream>>>(dst, cnt);
  inv_count<<<(NN + 255) / 256, 256, 0, stream>>>(cnt);

  // layer 1
  scatter_add<<<(NE * 32) / 256, 256, 0, stream>>>(x, src, dst, agg);
  sage_fused<<<NN / 16, 512, 0, stream>>>(x, agg, cnt, W1l, b1l, W1r, g1, be1, y1);

  // layer 2
  zero_f32<<<(ND + 255) / 256, 256, 0, stream>>>(agg, ND);
  scatter_add<<<(NE * 32) / 256, 256, 0, stream>>>(y1, src, dst, agg);
  sage_fused<<<NN / 16, 512, 0, stream>>>(y1, agg, cnt, W2l, b2l, W2r, g2, be2, out);
}